// AttentionSimilarityLinear_75617194213720
// MI455X (gfx1250) — compile-verified
//
#include <hip/hip_runtime.h>
#include <cstdint>

typedef float v2f __attribute__((ext_vector_type(2)));
typedef float v8f __attribute__((ext_vector_type(8)));

constexpr int B = 4, S = 8192, D = 1024, C = 256;
constexpr float EPS = 1e-8f;

// LDS row strides (floats): chosen for 16B alignment (async b128) and
// bank-conflict-free WMMA fragment reads (see analysis).
constexpr int XPAD = 36;   // 64x32 A-chunk
constexpr int WPAD = 264;  // 32x256 B-chunk

static __device__ __forceinline__ v8f wmma_f32_16x16x4(v2f a, v2f b, v8f c) {
  // D = A(16x4,f32) * B(4x16,f32) + C(16x16,f32)
  return __builtin_amdgcn_wmma_f32_16x16x4_f32(false, a, false, b, (short)0, c,
                                               false, false);
}

static __device__ __forceinline__ v8f zero8() {
  v8f z;
  for (int i = 0; i < 8; ++i) z[i] = 0.0f;
  return z;
}

// CDNA5 async memory->LDS copy, 16 bytes per lane. LDS dest is the wave-
// relative byte offset (low 32 bits of a generic pointer to __shared__).
// Tracked by ASYNCcnt; retire with s_wait_asynccnt + barrier.
static __device__ __forceinline__ void async_copy_b128(float* lds_dst,
                                                       const float* gsrc) {
  const uint32_t lds_addr = (uint32_t)(uintptr_t)lds_dst;
  asm volatile("global_load_async_to_lds_b128 %0, %1, off"
               :
               : "v"(lds_addr), "v"((uint64_t)(uintptr_t)gsrc)
               : "memory");
}

static __device__ __forceinline__ void wait_async0() {
  asm volatile("s_wait_asynccnt 0x0" ::: "memory");
}

// phi(x) = elu(x) + 1  ( = x+1 for x>=0, exp(x) for x<0 )
static __device__ __forceinline__ float phi_map(float x) {
  return x > 0.0f ? x + 1.0f : __expf(x);
}

// ---------------------------------------------------------------------------
// Zero small accumulators (K_sum: B*C, accG: B)
// ---------------------------------------------------------------------------
__global__ void zero_kernel(float* __restrict__ Ksum, float* __restrict__ accG) {
  int t = blockIdx.x * blockDim.x + threadIdx.x;
  if (t < B * C) Ksum[t] = 0.0f;
  if (t < B) accG[t] = 0.0f;
}

// ---------------------------------------------------------------------------
// out[b,s,c] = phi( X[b,s,:] @ W[:,c] ) * mask[b,s]
// grid (S/64, B), block 256 (8 waves). Block computes a 64x256 output tile.
// Wave w: row-tile (w&3), col-tiles ((w>>2)*8 + j), j=0..7.
// Both operands staged in LDS via async b128 (ASYNCcnt), WMMA-fed from LDS.
// ---------------------------------------------------------------------------
__global__ __launch_bounds__(256) void phi_proj_kernel(
    const float* __restrict__ X, const float* __restrict__ W,
    const float* __restrict__ mask, float* __restrict__ out) {
  __shared__ float ldsX[64 * XPAD];  // 9.0 KB
  __shared__ float ldsW[32 * WPAD];  // 33.0 KB
  const int tid = threadIdx.x;
  const int wave = tid >> 5;
  const int lane = tid & 31;
  const int hl = lane >> 4;   // lane half: selects K pair / row offset
  const int idx = lane & 15;  // M (A) or N (B/C/D) index
  const int rt = wave & 3;
  const int cg = wave >> 2;
  const int b = blockIdx.y;
  const int r0 = blockIdx.x * 64;

  v8f acc[8];
#pragma unroll
  for (int j = 0; j < 8; ++j) acc[j] = zero8();

  for (int kk = 0; kk < D; kk += 32) {
    // stage X[b][r0..r0+63][kk..kk+31]: 512 x b128, 2 per thread
#pragma unroll
    for (int i = 0; i < 2; ++i) {
      int e = i * 256 + tid;
      int row = e >> 3, c4 = e & 7;
      async_copy_b128(&ldsX[row * XPAD + c4 * 4],
                      &X[(size_t)(b * S + r0 + row) * D + kk + c4 * 4]);
    }
    // stage W[kk..kk+31][0..255]: 2048 x b128, 8 per thread
#pragma unroll
    for (int i = 0; i < 8; ++i) {
      int e = i * 256 + tid;
      int row = e >> 6, c4 = e & 63;
      async_copy_b128(&ldsW[row * WPAD + c4 * 4],
                      &W[(size_t)(kk + row) * C + c4 * 4]);
    }
    wait_async0();
    __syncthreads();
#pragma unroll
    for (int ks = 0; ks < 32; ks += 4) {
      const int k0 = ks + 2 * hl;
      v2f a;
      a.x = ldsX[(rt * 16 + idx) * XPAD + k0];
      a.y = ldsX[(rt * 16 + idx) * XPAD + k0 + 1];
#pragma unroll
      for (int j = 0; j < 8; ++j) {
        const int cn = (cg * 8 + j) * 16;
        v2f bf;
        bf.x = ldsW[k0 * WPAD + cn + idx];
        bf.y = ldsW[(k0 + 1) * WPAD + cn + idx];
        acc[j] = wmma_f32_16x16x4(a, bf, acc[j]);
      }
    }
    __syncthreads();  // protect LDS before next chunk's async writes
  }

  // epilogue: phi + mask, store
#pragma unroll
  for (int j = 0; j < 8; ++j) {
    const int c = (cg * 8 + j) * 16 + idx;
#pragma unroll
    for (int v = 0; v < 8; ++v) {
      const int s = r0 + rt * 16 + v + 8 * hl;
      const float val = phi_map(acc[j][v]) * mask[b * S + s];
      out[(size_t)(b * S + s) * C + c] = val;
    }
  }
}

// ---------------------------------------------------------------------------
// KV[b] = K[b]^T @ K[b]  (C x C). grid (32, B), block 256: one wave per
// 16x16 output tile (256 tiles/batch), K-loop over S in steps of 4.
// A[m][k] = K[s+k][cm0+m], B[k][n] = K[s+k][cn0+n] (both L2-resident).
// ---------------------------------------------------------------------------
__global__ __launch_bounds__(256) void kv_kernel(const float* __restrict__ K,
                                                 float* __restrict__ KV) {
  const int tid = threadIdx.x;
  const int wave = tid >> 5;
  const int lane = tid & 31;
  const int hl = lane >> 4;
  const int idx = lane & 15;
  const int b = blockIdx.y;
  const int tile = blockIdx.x * 8 + wave;  // 0..255
  const int cm0 = (tile >> 4) * 16;
  const int cn0 = (tile & 15) * 16;
  const float* Kb = K + (size_t)b * S * C;

  v8f acc = zero8();
#pragma unroll 4
  for (int s = 0; s < S; s += 4) {
    const int k0 = s + 2 * hl;
    v2f a, bf;
    a.x = Kb[(size_t)k0 * C + cm0 + idx];
    a.y = Kb[(size_t)(k0 + 1) * C + cm0 + idx];
    bf.x = Kb[(size_t)k0 * C + cn0 + idx];
    bf.y = Kb[(size_t)(k0 + 1) * C + cn0 + idx];
    acc = wmma_f32_16x16x4(a, bf, acc);
  }
#pragma unroll
  for (int v = 0; v < 8; ++v)
    KV[((size_t)b * C + cm0 + v + 8 * hl) * C + cn0 + idx] = acc[v];
}

// ---------------------------------------------------------------------------
// K_sum[b,c] = sum_s K[b,s,c]. Split-S, one global atomic per (block, c).
// grid (S/512, B), block 256 (thread t owns channel c=t).
// ---------------------------------------------------------------------------
__global__ __launch_bounds__(256) void ksum_kernel(const float* __restrict__ K,
                                                   float* __restrict__ Ksum) {
  const int c = threadIdx.x;
  const int b = blockIdx.y;
  const int s0 = blockIdx.x * 512;
  float p = 0.0f;
  for (int i = 0; i < 512; ++i) p += K[(size_t)(b * S + s0 + i) * C + c];
  atomicAdd(&Ksum[b * C + c], p);
}

// ---------------------------------------------------------------------------
// T = Q @ KV (WMMA), fused epilogue:
//   num_r = sum_c T[r,c]*Q[r,c],  den_r = sum_c Q[r,c]*K_sum[c]
//   row_l2 = max(num,eps)/max((den+eps)^2,eps); masked block-sum -> accG[b]
// grid (S/64, B), block 256. Q and KV chunks async-staged in LDS.
// ---------------------------------------------------------------------------
__global__ __launch_bounds__(256) void qkvq_kernel(
    const float* __restrict__ Q, const float* __restrict__ KV,
    const float* __restrict__ Ksum, const float* __restrict__ maskQ,
    float* __restrict__ accG) {
  __shared__ float ldsQ[64 * XPAD];
  __shared__ float ldsKV[32 * WPAD];
  __shared__ float red_num[64];
  __shared__ float red_den[64];
  __shared__ float blocksum;
  const int tid = threadIdx.x;
  const int wave = tid >> 5;
  const int lane = tid & 31;
  const int hl = lane >> 4;
  const int idx = lane & 15;
  const int rt = wave & 3;
  const int cg = wave >> 2;
  const int b = blockIdx.y;
  const int r0 = blockIdx.x * 64;

  if (tid < 64) {
    red_num[tid] = 0.0f;
    red_den[tid] = 0.0f;
  }
  if (tid == 0) blocksum = 0.0f;

  v8f acc[8];
#pragma unroll
  for (int j = 0; j < 8; ++j) acc[j] = zero8();

  for (int kc = 0; kc < C; kc += 32) {
#pragma unroll
    for (int i = 0; i < 2; ++i) {
      int e = i * 256 + tid;
      int row = e >> 3, c4 = e & 7;
      async_copy_b128(&ldsQ[row * XPAD + c4 * 4],
                      &Q[(size_t)(b * S + r0 + row) * C + kc + c4 * 4]);
    }
#pragma unroll
    for (int i = 0; i < 8; ++i) {
      int e = i * 256 + tid;
      int row = e >> 6, c4 = e & 63;
      async_copy_b128(&ldsKV[row * WPAD + c4 * 4],
                      &KV[((size_t)b * C + kc + row) * C + c4 * 4]);
    }
    wait_async0();
    __syncthreads();
#pragma unroll
    for (int ks = 0; ks < 32; ks += 4) {
      const int k0 = ks + 2 * hl;
      v2f a;
      a.x = ldsQ[(rt * 16 + idx) * XPAD + k0];
      a.y = ldsQ[(rt * 16 + idx) * XPAD + k0 + 1];
#pragma unroll
      for (int j = 0; j < 8; ++j) {
        const int cn = (cg * 8 + j) * 16;
        v2f bf;
        bf.x = ldsKV[k0 * WPAD + cn + idx];
        bf.y = ldsKV[(k0 + 1) * WPAD + cn + idx];
        acc[j] = wmma_f32_16x16x4(a, bf, acc[j]);
      }
    }
    __syncthreads();
  }

  // per-row partial sums via LDS atomics
#pragma unroll
  for (int j = 0; j < 8; ++j) {
    const int c = (cg * 8 + j) * 16 + idx;
    const float ksv = Ksum[b * C + c];
#pragma unroll
    for (int v = 0; v < 8; ++v) {
      const int row = rt * 16 + v + 8 * hl;
      const int s = r0 + row;
      const float qv = Q[(size_t)(b * S + s) * C + c];
      atomicAdd(&red_num[row], acc[j][v] * qv);
      atomicAdd(&red_den[row], qv * ksv);
    }
  }
  __syncthreads();

  if (tid < 64) {
    const int s = r0 + tid;
    const float num = fmaxf(red_num[tid], EPS);
    float den = red_den[tid] + EPS;
    den = fmaxf(den * den, EPS);
    const float val = (num / den) * maskQ[b * S + s];
    atomicAdd(&blocksum, val);
  }
  __syncthreads();
  if (tid == 0) atomicAdd(&accG[b], blocksum);
}

// ---------------------------------------------------------------------------
// Final: tau[b] = clip((n_keys*mean_l2 - 1)/max(n_keys-1,1e-6), 0, 1)
// ---------------------------------------------------------------------------
__global__ __launch_bounds__(256) void finalize_kernel(
    const float* __restrict__ maskQ, const float* __restrict__ maskK,
    const float* __restrict__ accG, float* __restrict__ out) {
  __shared__ float s1[256];
  __shared__ float s2[256];
  const int t = threadIdx.x;
  for (int b = 0; b < B; ++b) {
    float p1 = 0.0f, p2 = 0.0f;
    for (int s = t; s < S; s += 256) {
      p1 += maskQ[b * S + s];
      p2 += maskK[b * S + s];
    }
    s1[t] = p1;
    s2[t] = p2;
    __syncthreads();
    for (int off = 128; off > 0; off >>= 1) {
      if (t < off) {
        s1[t] += s1[t + off];
        s2[t] += s2[t + off];
      }
      __syncthreads();
    }
    if (t == 0) {
      const float qc = fmaxf(s1[0], 1.0f);
      const float nk = fmaxf(s2[0], 1.0f);
      const float mean_l2 = accG[b] / qc;
      const float tau = (nk * mean_l2 - 1.0f) / fmaxf(nk - 1.0f, 1e-6f);
      out[b] = fminf(fmaxf(tau, 0.0f), 1.0f);
    }
    __syncthreads();
  }
}

// ---------------------------------------------------------------------------
extern "C" void kernel_launch(void* const* d_in, const int* in_sizes, int n_in,
                              void* d_out, int out_size, void* d_ws,
                              size_t ws_size, hipStream_t stream) {
  (void)in_sizes; (void)n_in; (void)out_size; (void)ws_size;
  const float* Xc = (const float*)d_in[0];  // chosen_hidden_states (B,S,D)
  const float* Xr = (const float*)d_in[1];  // rejected_hidden_states (B,S,D)
  const float* mQ = (const float*)d_in[2];  // chosen_mask (B,S)
  const float* mK = (const float*)d_in[3];  // rejected_mask (B,S)
  const float* WQ = (const float*)d_in[4];  // (D,C)
  const float* WK = (const float*)d_in[5];  // (D,C)
  float* out = (float*)d_out;               // (B,)

  // Workspace layout (floats). K buffer is reused for Q after KV/K_sum.
  float* buf = (float*)d_ws;                       // B*S*C  (32 MB)
  float* KVb = buf + (size_t)B * S * C;            // B*C*C  (1 MB)
  float* Ksum = KVb + (size_t)B * C * C;           // B*C
  float* accG = Ksum + (size_t)B * C;              // B

  zero_kernel<<<dim3((B * C + 255) / 256), 256, 0, stream>>>(Ksum, accG);
  // K = phi(Xr @ W_K) * mask_K
  phi_proj_kernel<<<dim3(S / 64, B), 256, 0, stream>>>(Xr, WK, mK, buf);
  // KV = K^T K ; K_sum = sum_s K
  kv_kernel<<<dim3(32, B), 256, 0, stream>>>(buf, KVb);
  ksum_kernel<<<dim3(S / 512, B), 256, 0, stream>>>(buf, Ksum);
  // Q = phi(Xc @ W_Q) * mask_Q  (overwrites K buffer)
  phi_proj_kernel<<<dim3(S / 64, B), 256, 0, stream>>>(Xc, WQ, mQ, buf);
  // T = Q @ KV, fused row_l2 reduction into accG
  qkvq_kernel<<<dim3(S / 64, B), 256, 0, stream>>>(buf, KVb, Ksum, mQ, accG);
  finalize_kernel<<<1, 256, 0, stream>>>(mQ, mK, accG, out);
}